// LDW_down_38766374813928
// MI455X (gfx1250) — compile-verified
//
#include <hip/hip_runtime.h>
#include <hip/hip_bf16.h>

// Single-level separable 2-tap DWT, fully memory-bound (AI ~= 0.75 flop/byte).
// Strategy for MI455X: pure streaming kernel. wave32, 16B vector loads/stores,
// non-temporal cache hints (2.15 GB streamed once; don't pollute the 192MB L2),
// 4-deep unroll for MLP. WMMA intentionally unused: matrix formulation cannot
// beat the 23.3 TB/s bandwidth floor (~92 us) and only adds layout traffic.

typedef float v4f __attribute__((ext_vector_type(4)));
typedef float v2f __attribute__((ext_vector_type(2)));

__global__ __launch_bounds__(256) void LDW_down_dwt2d_kernel(
    const float* __restrict__ x,
    const float* __restrict__ lpf_h, const float* __restrict__ hpf_h,
    const float* __restrict__ lpf_v, const float* __restrict__ hpf_v,
    float* __restrict__ out)
{
    // Shapes: x (16,64,512,512), out (16,256,256,256) = [ll|hl|lh|hh] on C.
    // Thread decomposition: tid -> { wg: 4 out cols, rq: 4 out rows, c, b }.
    const unsigned tid = blockIdx.x * blockDim.x + threadIdx.x;
    const unsigned wg = tid & 63u;          // column group: out cols wg*4..+3
    const unsigned rq = (tid >> 6) & 63u;   // row quad:    out rows rq*4..+3
    const unsigned c  = (tid >> 12) & 63u;  // input channel
    const unsigned b  = tid >> 18;          // batch

    // Filter taps (uniform, cache-resident; default temporal policy).
    const v2f LH = *(const v2f*)lpf_h;
    const v2f HH = *(const v2f*)hpf_h;
    const v2f LV = *(const v2f*)lpf_v;
    const v2f HV = *(const v2f*)hpf_v;
    const float lh0 = LH.x, lh1 = LH.y;
    const float hh0 = HH.x, hh1 = HH.y;
    const float lv0 = LV.x, lv1 = LV.y;
    const float hv0 = HV.x, hv1 = HV.y;

    const float* plane = x + ((size_t)(b * 64u + c) << 18);            // 512*512 plane
    float*       obase = out + ((size_t)(b * 256u + c) << 16)          // ll channel plane
                             + (size_t)wg * 4u;                        // column offset
    const size_t SUB = (size_t)64u << 16;                              // 64 channels * 256*256

#pragma unroll
    for (int i = 0; i < 4; ++i) {
        const unsigned r = rq * 4u + (unsigned)i;                      // output row
        const float* row0 = plane + ((size_t)(2u * r) << 9) + (size_t)wg * 8u;
        const float* row1 = row0 + 512;

        // 2 rows x 8 consecutive input floats: 4x global_load_b128 (NT).
        v4f a0 = __builtin_nontemporal_load((const v4f*)(row0));
        v4f a1 = __builtin_nontemporal_load((const v4f*)(row0 + 4));
        v4f c0 = __builtin_nontemporal_load((const v4f*)(row1));
        v4f c1 = __builtin_nontemporal_load((const v4f*)(row1 + 4));

        // De-interleave even/odd columns.
        v4f e0 = __builtin_shufflevector(a0, a1, 0, 2, 4, 6);
        v4f o0 = __builtin_shufflevector(a0, a1, 1, 3, 5, 7);
        v4f e1 = __builtin_shufflevector(c0, c1, 0, 2, 4, 6);
        v4f o1 = __builtin_shufflevector(c0, c1, 1, 3, 5, 7);

        // Horizontal low/high on both rows.
        v4f l0 = e0 * lh0 + o0 * lh1;   // row 2r,   lowpass
        v4f h0 = e0 * hh0 + o0 * hh1;   // row 2r,   highpass
        v4f l1 = e1 * lh0 + o1 * lh1;   // row 2r+1, lowpass
        v4f h1 = e1 * hh0 + o1 * hh1;   // row 2r+1, highpass

        // Vertical low/high -> four subbands.
        v4f ll = l0 * lv0 + l1 * lv1;
        v4f hl = l0 * hv0 + l1 * hv1;
        v4f lh = h0 * lv0 + h1 * lv1;
        v4f hh = h0 * hv0 + h1 * hv1;

        // 4x global_store_b128 (NT), contiguous across lanes of the wave.
        float* orow = obase + ((size_t)r << 8);
        __builtin_nontemporal_store(ll, (v4f*)(orow));
        __builtin_nontemporal_store(hl, (v4f*)(orow + SUB));
        __builtin_nontemporal_store(lh, (v4f*)(orow + 2 * SUB));
        __builtin_nontemporal_store(hh, (v4f*)(orow + 3 * SUB));
    }
}

extern "C" void kernel_launch(void* const* d_in, const int* in_sizes, int n_in,
                              void* d_out, int out_size, void* d_ws, size_t ws_size,
                              hipStream_t stream) {
    (void)in_sizes; (void)n_in; (void)out_size; (void)d_ws; (void)ws_size;
    const float* x     = (const float*)d_in[0];
    const float* lpf_h = (const float*)d_in[1];
    const float* hpf_h = (const float*)d_in[2];
    const float* lpf_v = (const float*)d_in[3];
    const float* hpf_v = (const float*)d_in[4];
    float* out = (float*)d_out;

    // Total threads: 16 (b) * 64 (c) * 64 (row quads) * 64 (col groups) = 4,194,304
    const unsigned total_threads = 16u * 64u * 64u * 64u;
    const unsigned block = 256u;
    const unsigned grid = total_threads / block;   // 16384 blocks
    LDW_down_dwt2d_kernel<<<grid, block, 0, stream>>>(x, lpf_h, hpf_h, lpf_v, hpf_v, out);
}